// QPLayer_37108517438003
// MI455X (gfx1250) — compile-verified
//
#include <hip/hip_runtime.h>
#include <math.h>

// ---------------------------------------------------------------------------
// Batched simplex-constrained QP (FISTA) for MI455X / gfx1250.
//   minimize w^T Sigma w - mu^T w   s.t. sum w = 1, w >= 0
// B = 128 independent problems, N = 512.
// One workgroup (8 wave32) per batch. Matvecs via V_WMMA_F32_16X16X4_F32.
// Sigma working set (128 MB) is L2-resident (192 MB L2) -> L2-BW bound.
// Each wave owns 4 row-tiles with 4 live accumulators: one B fragment from
// LDS feeds 4 independent WMMAs per k-chunk (no D->C serial chain, 4x less
// DS traffic).
// ---------------------------------------------------------------------------

typedef float v2f __attribute__((ext_vector_type(2)));
typedef float v8f __attribute__((ext_vector_type(8)));

#define NN          512
#define BLOCK       256
#define NWAVES      (BLOCK / 32)
#define PWR_ITERS   30
#define FISTA_ITERS 500
#define PROJ_ITERS  16   // Michelot fixed-point steps (exact fixed point)

__device__ __forceinline__ float wave_allreduce_sum(float x) {
#pragma unroll
  for (int m = 16; m >= 1; m >>= 1) x += __shfl_xor(x, m, 32);
  return x;
}

// g = Sigma * y  (Sigma row-major NN x NN in global, y/g in LDS).
// Wave `wave` owns row-tiles {wave, wave+8, wave+16, wave+24}; K consumed 4
// at a time by f32 WMMA.
// A frag (16x4 f32): lane l holds S[row0 + (l&15)][k0 + 2*(l>>4) + {0,1}].
// B frag: y[k0+k] broadcast across all 16 columns (same per-half K mapping).
// D (16x16 f32): column n==0 (lanes 0 and 16) carries the matvec result.
__device__ __forceinline__ void matvec_wmma(const float* __restrict__ S,
                                            const float* __restrict__ y,
                                            float* __restrict__ g,
                                            int wave, int lane) {
  const int m  = lane & 15;
  const int kh = lane >> 4;  // half: K offset base 2*kh
  const float* rowp = S + (size_t)(wave * 16 + m) * NN + 2 * kh;

  v8f acc0 = {0.f, 0.f, 0.f, 0.f, 0.f, 0.f, 0.f, 0.f};
  v8f acc1 = acc0, acc2 = acc0, acc3 = acc0;

#pragma unroll 4
  for (int k0 = 0; k0 < NN; k0 += 4) {
    v2f bf = *(const v2f*)(y + k0 + 2 * kh);            // ds_load_b64 (shared)
    v2f a0 = *(const v2f*)(rowp + k0);                  // tile wave
    v2f a1 = *(const v2f*)(rowp + k0 + 128 * NN);       // tile wave+8
    v2f a2 = *(const v2f*)(rowp + k0 + 256 * NN);       // tile wave+16
    v2f a3 = *(const v2f*)(rowp + k0 + 384 * NN);       // tile wave+24
    acc0 = __builtin_amdgcn_wmma_f32_16x16x4_f32(false, a0, false, bf,
                                                 (short)0, acc0, false, false);
    acc1 = __builtin_amdgcn_wmma_f32_16x16x4_f32(false, a1, false, bf,
                                                 (short)0, acc1, false, false);
    acc2 = __builtin_amdgcn_wmma_f32_16x16x4_f32(false, a2, false, bf,
                                                 (short)0, acc2, false, false);
    acc3 = __builtin_amdgcn_wmma_f32_16x16x4_f32(false, a3, false, bf,
                                                 (short)0, acc3, false, false);
  }

  if (m == 0) {  // lanes 0 (M=0..7) and 16 (M=8..15) hold column n=0
    const int rbase = wave * 16 + 8 * kh;
#pragma unroll
    for (int i = 0; i < 4; ++i) {
      const v8f acc = (i == 0) ? acc0 : (i == 1) ? acc1 : (i == 2) ? acc2
                                                                   : acc3;
      float* gp = g + rbase + i * 128;
      *(float4*)(gp)     = make_float4(acc[0], acc[1], acc[2], acc[3]);
      *(float4*)(gp + 4) = make_float4(acc[4], acc[5], acc[6], acc[7]);
    }
  }
}

__global__ void __launch_bounds__(BLOCK)
qp_fista_kernel(const float* __restrict__ mu_g,
                const float* __restrict__ Sigma_g,
                float* __restrict__ out) {
  __shared__ __align__(16) float yv[NN];   // current y / power-iteration b
  __shared__ __align__(16) float gv[NN];   // matvec result
  __shared__ __align__(16) float vv[NN];   // pre-projection point
  __shared__ float red[NWAVES];

  const int b    = blockIdx.x;
  const int tid  = threadIdx.x;
  const int lane = tid & 31;
  const int wave = tid >> 5;
  const int i0 = tid, i1 = tid + BLOCK;

  const float* S  = Sigma_g + (size_t)b * NN * NN;
  const float* mb = mu_g + (size_t)b * NN;
  const float mu_a = mb[i0];
  const float mu_b = mb[i1];

  // ---- Power iteration for Lipschitz constant: b <- Sb / (||Sb|| + 1e-12)
  yv[i0] = 1.0f / NN;
  yv[i1] = 1.0f / NN;
  __syncthreads();

  for (int it = 0; it < PWR_ITERS; ++it) {
    matvec_wmma(S, yv, gv, wave, lane);
    __syncthreads();  // gv ready
    float p = gv[i0] * gv[i0] + gv[i1] * gv[i1];
    p = wave_allreduce_sum(p);
    if (lane == 0) red[wave] = p;
    __syncthreads();  // red ready
    float nrm2 = 0.f;
#pragma unroll
    for (int w = 0; w < NWAVES; ++w) nrm2 += red[w];
    const float inv = 1.0f / (sqrtf(nrm2) + 1e-12f);
    yv[i0] = gv[i0] * inv;
    yv[i1] = gv[i1] * inv;
    __syncthreads();  // yv ready for next matvec
  }

  // ---- lmax = b^T Sigma b ; step = 1/(2*lmax + 1e-8)
  matvec_wmma(S, yv, gv, wave, lane);
  __syncthreads();
  {
    float p = yv[i0] * gv[i0] + yv[i1] * gv[i1];
    p = wave_allreduce_sum(p);
    if (lane == 0) red[wave] = p;
  }
  __syncthreads();
  float lmax = 0.f;
#pragma unroll
  for (int w = 0; w < NWAVES; ++w) lmax += red[w];
  const float step = 1.0f / (2.0f * lmax + 1e-8f);

  // ---- FISTA on the simplex
  float w_a = 1.0f / NN, w_b = 1.0f / NN;  // w0
  float t = 1.0f;
  __syncthreads();
  yv[i0] = w_a;
  yv[i1] = w_b;
  __syncthreads();

  for (int it = 0; it < FISTA_ITERS; ++it) {
    matvec_wmma(S, yv, gv, wave, lane);   // gv = Sigma * y
    __syncthreads();                      // gv ready

    const float v_a = yv[i0] - step * (2.0f * gv[i0] - mu_a);
    const float v_b = yv[i1] - step * (2.0f * gv[i1] - mu_b);
    vv[i0] = v_a;
    vv[i1] = v_b;
    __syncthreads();                      // vv ready

    // Wave-redundant simplex projection: Michelot fixed point for theta.
    // Each lane keeps 16 of the 512 elements in registers; reductions are
    // pure shfl butterflies -> no barriers inside the projection loop.
    float r[16];
    const float4* vp = (const float4*)(vv + lane * 16);
#pragma unroll
    for (int q = 0; q < 4; ++q) {
      float4 f = vp[q];
      r[4 * q + 0] = f.x; r[4 * q + 1] = f.y;
      r[4 * q + 2] = f.z; r[4 * q + 3] = f.w;
    }
    float s = 0.f;
#pragma unroll
    for (int j = 0; j < 16; ++j) s += r[j];
    s = wave_allreduce_sum(s);
    float theta = (s - 1.0f) / (float)NN;
#pragma unroll 1
    for (int pi = 0; pi < PROJ_ITERS; ++pi) {
      float ps = 0.f, pc = 0.f;
#pragma unroll
      for (int j = 0; j < 16; ++j) {
        if (r[j] > theta) { ps += r[j]; pc += 1.0f; }
      }
      ps = wave_allreduce_sum(ps);
      pc = wave_allreduce_sum(pc);
      theta = (ps - 1.0f) / fmaxf(pc, 1.0f);
    }

    const float wn_a = fmaxf(v_a - theta, 0.f);
    const float wn_b = fmaxf(v_b - theta, 0.f);
    const float tn   = 0.5f * (1.0f + sqrtf(1.0f + 4.0f * t * t));
    const float beta = (t - 1.0f) / tn;
    yv[i0] = wn_a + beta * (wn_a - w_a);
    yv[i1] = wn_b + beta * (wn_b - w_b);
    w_a = wn_a; w_b = wn_b; t = tn;
    __syncthreads();                      // yv ready for next iteration
  }

  out[(size_t)b * NN + i0] = w_a;
  out[(size_t)b * NN + i1] = w_b;
}

extern "C" void kernel_launch(void* const* d_in, const int* in_sizes, int n_in,
                              void* d_out, int out_size, void* d_ws, size_t ws_size,
                              hipStream_t stream) {
  (void)n_in; (void)out_size; (void)d_ws; (void)ws_size;
  const float* mu    = (const float*)d_in[0];   // [B, N] f32
  const float* Sigma = (const float*)d_in[1];   // [B, N, N] f32 (symmetric)
  float* out = (float*)d_out;                   // [B, N] f32
  const int B = in_sizes[0] / NN;
  qp_fista_kernel<<<B, BLOCK, 0, stream>>>(mu, Sigma, out);
}